// NodeRegressor_80023830659433
// MI455X (gfx1250) — compile-verified
//
#include <hip/hip_runtime.h>

// ---------------------------------------------------------------------------
// GraphSAGE 2-layer + linear head for MI455X (gfx1250, wave32, WMMA).
//   layer: out = relu( mean_agg(x) @ Wl.T + b + x @ Wr.T )
// Scatter phase: fp32 HW atomics into L2-resident aggregation buffers,
//                lane-per-(edge,quad) for coalesced gathers/atomics.
// Dense phase:   V_WMMA_F32_16X16X4_F32, one wave = 16-node x 64-col tile.
// ---------------------------------------------------------------------------

typedef __attribute__((ext_vector_type(2))) float v2f;
typedef __attribute__((ext_vector_type(8))) float v8f;

#define N_NODES 100000
#define IN_C    32
#define HID_C   64
#define OUT_C   5
#define N_EDGES 1600000

__device__ __forceinline__ v8f wmma_f32(v2f a, v2f b, v8f c) {
    // 8 args: (neg_a, A, neg_b, B, c_mod, C, reuse_a, reuse_b)
    return __builtin_amdgcn_wmma_f32_16x16x4_f32(false, a, false, b,
                                                 (short)0, c, false, false);
}

__device__ __forceinline__ void fadd_atomic(float* p, float v) {
#if defined(__HIP_DEVICE_COMPILE__)
    unsafeAtomicAdd(p, v);   // lowers to global_atomic_add_f32
#else
    (void)p; (void)v;
#endif
}

// ---------------- scatter: layer 1 (32 ch, 8 quads/edge) + counts ----------
// thread t -> edge t>>3, quad t&7 : 8 lanes stream one node row (128B burst)
__global__ __launch_bounds__(256) void scatter1_kernel(
    const float* __restrict__ x, const int* __restrict__ ei,
    float* __restrict__ agg, float* __restrict__ cnt)
{
    const long long t = (long long)blockIdx.x * blockDim.x + threadIdx.x;
    const int e = (int)(t >> 3);
    const int q = (int)(t & 7);
    if (e >= N_EDGES) return;
    const int src = ei[e];
    const int dst = ei[N_EDGES + e];
    if (q == 0) fadd_atomic(&cnt[dst], 1.0f);
    const float4 v = *(const float4*)(x + (size_t)src * IN_C + 4 * q);
    float* a = agg + (size_t)dst * IN_C + 4 * q;
    fadd_atomic(a + 0, v.x);
    fadd_atomic(a + 1, v.y);
    fadd_atomic(a + 2, v.z);
    fadd_atomic(a + 3, v.w);
}

// ---------------- scatter: layer 2 (64 ch, 16 quads/edge) ------------------
__global__ __launch_bounds__(256) void scatter2_kernel(
    const float* __restrict__ h, const int* __restrict__ ei,
    float* __restrict__ agg)
{
    const long long t = (long long)blockIdx.x * blockDim.x + threadIdx.x;
    const int e = (int)(t >> 4);
    const int q = (int)(t & 15);
    if (e >= N_EDGES) return;
    const int src = ei[e];
    const int dst = ei[N_EDGES + e];
    const float4 v = *(const float4*)(h + (size_t)src * HID_C + 4 * q);
    float* a = agg + (size_t)dst * HID_C + 4 * q;
    fadd_atomic(a + 0, v.x);
    fadd_atomic(a + 1, v.y);
    fadd_atomic(a + 2, v.z);
    fadd_atomic(a + 3, v.w);
}

// ---------------- WMMA SAGE layer (templated on K) -------------------------
// One wave computes h[tile*16 .. +16, 0..63] = relu(mean@Wl.T + b + xin@Wr.T)
template <int K>
__global__ __launch_bounds__(256) void sage_layer_kernel(
    const float* __restrict__ xin,   // [N,K]  self features
    const float* __restrict__ agg,   // [N,K]  summed neighbor features
    const float* __restrict__ cnt,   // [N]
    const float* __restrict__ Wl,    // [64,K]
    const float* __restrict__ bl,    // [64]
    const float* __restrict__ Wr,    // [64,K]
    float* __restrict__ hout)        // [N,64]
{
    const int lane = threadIdx.x & 31;
    const int wave = threadIdx.x >> 5;
    const int tile = blockIdx.x * (blockDim.x >> 5) + wave;
    const int nTiles = N_NODES / 16;
    if (tile >= nTiles) return;          // wave-uniform: EXEC stays all-1s

    const int r    = lane & 15;          // A-row / B-col / D-col index
    const int koff = (lane >> 4) * 2;    // K sub-offset: 0 or 2
    const int row0 = tile * 16;
    const int row  = row0 + r;

    const float inv = 1.0f / fmaxf(cnt[row], 1.0f);
    const float* am_p = agg + (size_t)row * K;
    const float* ax_p = xin + (size_t)row * K;

    v8f acc[4];
#pragma unroll
    for (int nt = 0; nt < 4; ++nt) { v8f z = {}; acc[nt] = z; }

#pragma unroll
    for (int kk = 0; kk < K / 4; ++kk) {
        const int k0 = kk * 4 + koff;
        v2f am; am.x = am_p[k0] * inv; am.y = am_p[k0 + 1] * inv;
        v2f ax; ax.x = ax_p[k0];       ax.y = ax_p[k0 + 1];
#pragma unroll
        for (int nt = 0; nt < 4; ++nt) {
            const int n = nt * 16 + r;
            v2f bL; bL.x = Wl[n * K + k0]; bL.y = Wl[n * K + k0 + 1];
            v2f bR; bR.x = Wr[n * K + k0]; bR.y = Wr[n * K + k0 + 1];
            acc[nt] = wmma_f32(am, bL, acc[nt]);
            acc[nt] = wmma_f32(ax, bR, acc[nt]);
        }
    }

    // D layout: VGPR vr -> row vr (lanes 0-15) / vr+8 (lanes 16-31); col = r
    const int rAdd = (lane >> 4) * 8;
#pragma unroll
    for (int nt = 0; nt < 4; ++nt) {
        const int col  = nt * 16 + r;
        const float bb = bl[col];
#pragma unroll
        for (int vr = 0; vr < 8; ++vr) {
            const int node = row0 + vr + rAdd;
            hout[(size_t)node * HID_C + col] = fmaxf(acc[nt][vr] + bb, 0.0f);
        }
    }
}

// ---------------- WMMA final linear: [N,64] @ [64,5].T + b -----------------
// Branchless: clamp the weight row for lanes with col >= 5; the resulting
// garbage output columns 5..15 are computed but never stored.
__global__ __launch_bounds__(256) void final_lin_kernel(
    const float* __restrict__ h,     // [N,64]
    const float* __restrict__ Wlin,  // [5,64]
    const float* __restrict__ blin,  // [5]
    float* __restrict__ out)         // [N,5]
{
    const int lane = threadIdx.x & 31;
    const int wave = threadIdx.x >> 5;
    const int tile = blockIdx.x * (blockDim.x >> 5) + wave;
    const int nTiles = N_NODES / 16;
    if (tile >= nTiles) return;

    const int r    = lane & 15;
    const int koff = (lane >> 4) * 2;
    const int row0 = tile * 16;
    const int row  = row0 + r;
    const float* hp = h + (size_t)row * HID_C;
    const int rw = (r < OUT_C) ? r : 0;              // clamped, in-bounds
    const float* wp = Wlin + (size_t)rw * HID_C;

    v8f acc = {};
#pragma unroll
    for (int kk = 0; kk < HID_C / 4; ++kk) {
        const int k0 = kk * 4 + koff;
        v2f a; a.x = hp[k0]; a.y = hp[k0 + 1];
        v2f b; b.x = wp[k0]; b.y = wp[k0 + 1];
        acc = wmma_f32(a, b, acc);
    }

    const int rAdd = (lane >> 4) * 8;
    const int col  = r;
    if (col < OUT_C) {
        const float bb = blin[col];
#pragma unroll
        for (int vr = 0; vr < 8; ++vr) {
            const int node = row0 + vr + rAdd;
            out[(size_t)node * OUT_C + col] = acc[vr] + bb;
        }
    }
}

// ---------------------------------------------------------------------------
extern "C" void kernel_launch(void* const* d_in, const int* in_sizes, int n_in,
                              void* d_out, int out_size, void* d_ws, size_t ws_size,
                              hipStream_t stream)
{
    (void)in_sizes; (void)n_in; (void)out_size; (void)ws_size;

    const float* x    = (const float*)d_in[0];
    const int*   ei   = (const int*)d_in[1];   // [2, E] int32
    const float* W1l  = (const float*)d_in[2];
    const float* b1   = (const float*)d_in[3];
    const float* W1r  = (const float*)d_in[4];
    const float* W2l  = (const float*)d_in[5];
    const float* b2   = (const float*)d_in[6];
    const float* W2r  = (const float*)d_in[7];
    const float* Wlin = (const float*)d_in[8];
    const float* blin = (const float*)d_in[9];
    float* out = (float*)d_out;

    // workspace layout (floats): cnt | agg1 | agg2 | h1 | h2
    float* ws   = (float*)d_ws;
    float* cnt  = ws;
    float* agg1 = cnt  + (size_t)N_NODES;
    float* agg2 = agg1 + (size_t)N_NODES * IN_C;
    float* h1   = agg2 + (size_t)N_NODES * HID_C;
    float* h2   = h1   + (size_t)N_NODES * HID_C;

    // zero cnt + agg1 + agg2 (contiguous)
    hipMemsetAsync(ws, 0, (size_t)N_NODES * (1 + IN_C + HID_C) * sizeof(float), stream);

    const int tB = 256;
    const long long s1Threads = (long long)N_EDGES * 8;   // 8 quads / edge
    const long long s2Threads = (long long)N_EDGES * 16;  // 16 quads / edge
    const int g1 = (int)((s1Threads + tB - 1) / tB);
    const int g2 = (int)((s2Threads + tB - 1) / tB);

    const int nTiles = N_NODES / 16;
    const int wavesPerBlock = 8;                          // 256 threads = 8 waves
    const int gT = (nTiles + wavesPerBlock - 1) / wavesPerBlock;

    scatter1_kernel<<<g1, tB, 0, stream>>>(x, ei, agg1, cnt);
    sage_layer_kernel<IN_C><<<gT, 256, 0, stream>>>(x, agg1, cnt, W1l, b1, W1r, h1);
    scatter2_kernel<<<g2, tB, 0, stream>>>(h1, ei, agg2);
    sage_layer_kernel<HID_C><<<gT, 256, 0, stream>>>(h1, agg2, cnt, W2l, b2, W2r, h2);
    final_lin_kernel<<<gT, 256, 0, stream>>>(h2, Wlin, blin, out);
}